// EncodePosition_9448928051745
// MI455X (gfx1250) — compile-verified
//
#include <hip/hip_runtime.h>
#include <math.h>
#include <stdint.h>

typedef float v2f __attribute__((ext_vector_type(2)));
typedef float v4f __attribute__((ext_vector_type(4)));
typedef float v8f __attribute__((ext_vector_type(8)));
typedef uint32_t u32x4 __attribute__((ext_vector_type(4)));
typedef uint32_t u32x8 __attribute__((ext_vector_type(8)));

#define BINS   16
#define NB     8
#define NPTS   2048
#define FDIM   128
#define HDIM   64
#define RPB    128     // points (rows) handled per block
#define NTHR   256     // 8 wave32s
#define NWAVE  8

// ---- LDS arena layout (float offsets) -------------------------------------
// [0,16384): phase1 = pts[2048][4]=(x,y,z,0) + sqn[2048]
//            phase3 = per-wave h1/h2 pair-major (8 waves x 2048 floats)
#define OFF_PTS   0       // 8192 floats
#define OFF_SQ    8192    // 2048 floats
#define OFF_HIST  16384   // hist[128][16]
#define OFF_HTP   18432   // histT pair-major: [8 pairs][128 cols] x float2
#define OFF_ST    20480   // s1[64] t1[64] s2[64] t2[64] (folded BN scale/shift)
#define OFF_RS    20736   // 1/rowsum[128]
#define OFF_B3    20864   // b3[128]
#define OFF_W1    20992   // w1: 64x16
#define OFF_W2    22016   // w2: 64x64
#define OFF_W3    26112   // w3: 128x64
#define SM_FLOATS 34304   // 137216 bytes (WGP has 320KB)

__device__ __forceinline__ v8f wmma_f32x4(v2f a, v2f b, v8f c) {
  // D = A(16x4 f32) * B(4x16 f32) + C(16x16 f32)
  return __builtin_amdgcn_wmma_f32_16x16x4_f32(false, a, false, b, (short)0, c,
                                               false, false);
}

// Issue one TDM 1-D copy: nfloats f32 from global gptr -> LDS byte offset.
// D# per CDNA5 ISA ch8: group0 = {count|..., lds_addr, global_addr, type=2},
// group1 = {data_size, tensor_dim0, tile_dim0, strides}.
__device__ __forceinline__ void tdm_load_f32(const void* gptr, uint32_t lds_byte,
                                             uint32_t nfloats) {
  uint64_t ga = (uint64_t)(uintptr_t)gptr;
  union { u32x4 v; uint32_t w[4]; } g0;
  union { u32x8 v; uint32_t w[8]; } g1;
  g0.w[0] = 1u;                                        // count=1, user mode
  g0.w[1] = lds_byte;                                  // lds_addr
  g0.w[2] = (uint32_t)ga;                              // global_addr[31:0]
  g0.w[3] = (uint32_t)((ga >> 32) & 0x1FFFFFFu) | (2u << 30);  // [56:32], type=2
  g1.w[0] = 2u << 16;                                  // data_size = 4 bytes
  g1.w[1] = (nfloats & 0xFFFFu) << 16;                 // tensor_dim0[15:0]
  g1.w[2] = (nfloats >> 16) & 0xFFFFu;                 // tensor_dim0[31:16]
  g1.w[3] = (nfloats & 0xFFFFu) << 16;                 // tile_dim0 = nfloats (<=65535)
  g1.w[4] = 0u;                                        // tile_dim1/2 unused
  g1.w[5] = nfloats;                                   // tensor_dim0_stride (benign)
  g1.w[6] = 0u;
  g1.w[7] = 0u;
  asm volatile("tensor_load_to_lds %0, %1" :: "s"(g0.v), "s"(g1.v) : "memory");
}

__global__ __launch_bounds__(NTHR)
void hist_mlp_fused(const float* __restrict__ xin, const float* __restrict__ fea,
                    const float* __restrict__ w1, const float* __restrict__ b1,
                    const float* __restrict__ g1, const float* __restrict__ be1,
                    const float* __restrict__ m1, const float* __restrict__ v1,
                    const float* __restrict__ w2, const float* __restrict__ b2,
                    const float* __restrict__ g2, const float* __restrict__ be2,
                    const float* __restrict__ m2, const float* __restrict__ v2,
                    const float* __restrict__ w3, const float* __restrict__ b3,
                    float* __restrict__ out) {
  __shared__ float sm[SM_FLOATS];

  const int tid     = threadIdx.x;
  const int bb      = blockIdx.x >> 4;          // batch
  const int rowBase = (blockIdx.x & 15) * RPB;  // first point this block owns
  const int wv      = tid >> 5;
  const int lane    = tid & 31;
  const int half    = lane >> 4;                // 0: lanes 0-15, 1: lanes 16-31
  const int idx     = lane & 15;

  // ---------------- phase 0: TDM weight stage + coords / BN fold -----------
  if (wv == 0) {  // one wave issues the async weight DMAs (overlap phase 1/2)
    tdm_load_f32(w1, OFF_W1 * 4u, HDIM * BINS);
    tdm_load_f32(w2, OFF_W2 * 4u, HDIM * HDIM);
    tdm_load_f32(w3, OFF_W3 * 4u, FDIM * HDIM);
  }
  {  // prefetch this block's fea slice (128 rows x 512B = 512 lines of 128B)
    const float* fbase = fea + (size_t)bb * FDIM * NPTS + rowBase;
    for (int l = tid; l < FDIM * 4; l += NTHR)
      __builtin_prefetch(fbase + (size_t)(l >> 2) * NPTS + (l & 3) * 32, 0, 1);
  }
  for (int n = tid; n < NPTS; n += NTHR) {
    const float* p = xin + ((size_t)bb * NPTS + n) * 3;
    float px = p[0], py = p[1], pz = p[2];
    v4f pk; pk.x = px; pk.y = py; pk.z = pz; pk.w = 0.0f;
    *reinterpret_cast<v4f*>(&sm[OFF_PTS + (n << 2)]) = pk;
    sm[OFF_SQ + n] = px * px + py * py + pz * pz;
  }
  for (int e = tid; e < RPB * BINS; e += NTHR) sm[OFF_HIST + e] = 0.0f;
  if (tid < HDIM) {
    float inv1 = g1[tid] * rsqrtf(v1[tid] + 1e-5f);
    sm[OFF_ST + tid]       = inv1;
    sm[OFF_ST + 64 + tid]  = b1[tid] * inv1 + be1[tid] - m1[tid] * inv1;
    float inv2 = g2[tid] * rsqrtf(v2[tid] + 1e-5f);
    sm[OFF_ST + 128 + tid] = inv2;
    sm[OFF_ST + 192 + tid] = b2[tid] * inv2 + be2[tid] - m2[tid] * inv2;
  }
  if (tid < FDIM) sm[OFF_B3 + tid] = b3[tid];
  __syncthreads();

  // ---------------- phase 1: pairwise distances via WMMA, histogram --------
  {
    const int row0 = rowBase + wv * 16;  // this wave's 16 rows
    // A fragment: rows row0..row0+15, K = (x,y) for lanes<16, (z,0) for >=16
    const v2f a =
        *reinterpret_cast<const v2f*>(&sm[OFF_PTS + ((row0 + idx) << 2) + 2 * half]);
    float sqrow[8];
#pragma unroll
    for (int v = 0; v < 8; ++v)
      sqrow[v] = sm[OFF_SQ + row0 + v + 8 * half];
    const int histBase = OFF_HIST + (wv * 16 + 8 * half) * BINS;

    for (int ct = 0; ct < NPTS / 16; ++ct) {
      const int col = ct * 16 + idx;
      const v2f bv =
          *reinterpret_cast<const v2f*>(&sm[OFF_PTS + (col << 2) + 2 * half]);
      const float sqc = sm[OFF_SQ + col];
      v8f c = {};
      c = wmma_f32x4(a, bv, c);  // c[v] = dot(row0+v+8*half, col)
#pragma unroll
      for (int v = 0; v < 8; ++v) {
        float d2  = sqrow[v] + sqc - 2.0f * c[v];
        float d   = sqrtf(fmaxf(d2, 0.0f));
        int bin   = (int)floorf((d - 1.0f) * (16.0f / 79.0f));
        bin       = bin < 0 ? 0 : (bin > 15 ? 15 : bin);
        float inc = (d >= 1.0f && d <= 80.0f) ? 1.0f : 0.0f;
        atomicAdd(&sm[histBase + v * BINS + bin], inc);  // branchless
      }
    }
  }
  __syncthreads();

  // ------------- phase 2: normalize + transpose to pair-major histT --------
  if (tid < RPB) {
    float s = 0.0f;
#pragma unroll
    for (int k = 0; k < BINS; ++k) s += sm[OFF_HIST + tid * BINS + k];
    sm[OFF_RS + tid] = 1.0f / s;   // reference divides; s==0 -> inf like ref
  }
  __syncthreads();
  for (int e = tid; e < RPB * BINS / 2; e += NTHR) {
    int r = e & 127, p = e >> 7;   // p = bin pair 0..7
    float rs = sm[OFF_RS + r];
    v2f hp;
    hp.x = sm[OFF_HIST + r * BINS + 2 * p]     * rs;
    hp.y = sm[OFF_HIST + r * BINS + 2 * p + 1] * rs;
    *reinterpret_cast<v2f*>(&sm[OFF_HTP + ((p << 7) + r) * 2]) = hp;
  }
  if (wv == 0) __builtin_amdgcn_s_wait_tensorcnt(0);  // weights landed in LDS
  __syncthreads();

  // ---------------- phase 3: fused MLP (16 -> 64 -> 64 -> 128) -------------
  // pair-major intermediates: hXp[pair p][col] = (h[2p][col], h[2p+1][col])
  float* h1p = &sm[wv * 2048];          // 32 pairs x 16 cols x float2
  float* h2p = h1p + 1024;
  const float* ST = &sm[OFF_ST];
  const int ncol  = wv * 16 + idx;      // this lane's local column (point)

  // conv1: h1 = relu(bn(W1[64x16] * hT))
#pragma unroll
  for (int rt = 0; rt < 4; ++rt) {
    v8f acc = {};
#pragma unroll
    for (int k0 = 0; k0 < BINS; k0 += 4) {
      const v2f a = *reinterpret_cast<const v2f*>(
          &sm[OFF_W1 + (rt * 16 + idx) * BINS + k0 + 2 * half]);
      const int p = (k0 >> 1) + half;
      const v2f bv =
          *reinterpret_cast<const v2f*>(&sm[OFF_HTP + ((p << 7) + ncol) * 2]);
      acc = wmma_f32x4(a, bv, acc);
    }
#pragma unroll
    for (int v = 0; v < 8; v += 2) {
      int row = rt * 16 + v + 8 * half;      // even row of the pair
      v2f r;
      r.x = fmaxf(acc[v]     * ST[row]     + ST[64 + row],     0.0f);
      r.y = fmaxf(acc[v + 1] * ST[row + 1] + ST[64 + row + 1], 0.0f);
      *reinterpret_cast<v2f*>(&h1p[((row >> 1) * 16 + idx) * 2]) = r;
    }
  }

  // conv2: h2 = relu(bn(W2[64x64] * h1))
#pragma unroll
  for (int rt = 0; rt < 4; ++rt) {
    v8f acc = {};
#pragma unroll
    for (int k0 = 0; k0 < HDIM; k0 += 4) {
      const v2f a = *reinterpret_cast<const v2f*>(
          &sm[OFF_W2 + (rt * 16 + idx) * HDIM + k0 + 2 * half]);
      const int p = (k0 >> 1) + half;
      const v2f bv = *reinterpret_cast<const v2f*>(&h1p[(p * 16 + idx) * 2]);
      acc = wmma_f32x4(a, bv, acc);
    }
#pragma unroll
    for (int v = 0; v < 8; v += 2) {
      int row = rt * 16 + v + 8 * half;
      v2f r;
      r.x = fmaxf(acc[v]     * ST[128 + row]     + ST[192 + row],     0.0f);
      r.y = fmaxf(acc[v + 1] * ST[128 + row + 1] + ST[192 + row + 1], 0.0f);
      *reinterpret_cast<v2f*>(&h2p[((row >> 1) * 16 + idx) * 2]) = r;
    }
  }

  // conv3: out = W3[128x64] * h2 + b3 + fea
  const int nglob = rowBase + wv * 16 + idx;
#pragma unroll
  for (int rt = 0; rt < 8; ++rt) {
    v8f acc = {};
#pragma unroll
    for (int k0 = 0; k0 < HDIM; k0 += 4) {
      const v2f a = *reinterpret_cast<const v2f*>(
          &sm[OFF_W3 + (rt * 16 + idx) * HDIM + k0 + 2 * half]);
      const int p = (k0 >> 1) + half;
      const v2f bv = *reinterpret_cast<const v2f*>(&h2p[(p * 16 + idx) * 2]);
      acc = wmma_f32x4(a, bv, acc);
    }
#pragma unroll
    for (int v = 0; v < 8; ++v) {
      int row = rt * 16 + v + 8 * half;
      size_t o = ((size_t)bb * FDIM + row) * NPTS + nglob;
      out[o] = acc[v] + sm[OFF_B3 + row] + fea[o];
    }
  }
}

extern "C" void kernel_launch(void* const* d_in, const int* in_sizes, int n_in,
                              void* d_out, int out_size, void* d_ws, size_t ws_size,
                              hipStream_t stream) {
  (void)in_sizes; (void)n_in; (void)out_size; (void)d_ws; (void)ws_size;
  const float* x   = (const float*)d_in[0];
  const float* fea = (const float*)d_in[1];
  const float* w1  = (const float*)d_in[2];
  const float* b1  = (const float*)d_in[3];
  const float* g1  = (const float*)d_in[4];
  const float* be1 = (const float*)d_in[5];
  const float* m1  = (const float*)d_in[6];
  const float* v1  = (const float*)d_in[7];
  const float* w2  = (const float*)d_in[8];
  const float* b2  = (const float*)d_in[9];
  const float* g2  = (const float*)d_in[10];
  const float* be2 = (const float*)d_in[11];
  const float* m2  = (const float*)d_in[12];
  const float* v2  = (const float*)d_in[13];
  const float* w3  = (const float*)d_in[14];
  const float* b3  = (const float*)d_in[15];
  float* out = (float*)d_out;

  dim3 grid(NB * (NPTS / RPB));  // 8 batches x 16 row tiles = 128 blocks
  dim3 block(NTHR);
  hist_mlp_fused<<<grid, block, 0, stream>>>(x, fea, w1, b1, g1, be1, m1, v1,
                                             w2, b2, g2, be2, m2, v2, w3, b3, out);
}